// ReciprocatorLM_18245021073731
// MI455X (gfx1250) — compile-verified
//
#include <hip/hip_runtime.h>

// ---------------------------------------------------------------------------
// Complex attention layer for MI455X (gfx1250, wave32, WMMA).
// fp32 tensor path: V_WMMA_F32_16X16X4_F32 (exact precision match with the
// complex64 reference).  Complex GEMM = 4 real WMMAs.
// All WMMA B-operands use pair-major LDS layouts (B[k/2][n][2]) so every
// fragment is one aligned ds_load_b64 -- no repacking moves.
// Layout-compatible stages (GEMM A-tiles, attention Q tile) use the CDNA5
// async global->LDS DMA path (ASYNCcnt) when the toolchain exposes it.
// ---------------------------------------------------------------------------

typedef __attribute__((ext_vector_type(2))) float v2f;
typedef __attribute__((ext_vector_type(4))) float f4v;
typedef __attribute__((ext_vector_type(8))) float v8f;
typedef __attribute__((ext_vector_type(4))) int   i4v;

namespace {
constexpr int Bn  = 2;
constexpr int Sn  = 2048;
constexpr int Hn  = 1024;
constexpr int Ln  = 2048;
constexpr int KVn = Ln + Sn;   // 4096
constexpr int NH  = 16;
constexpr int HD  = 64;
constexpr int Mn  = Bn * Sn;   // 4096 rows (b,s)
}

#if defined(__HIP_DEVICE_COMPILE__) && \
    __has_builtin(__builtin_amdgcn_global_load_async_to_lds_b128)
#define HAVE_ASYNC_LDS 1
#else
#define HAVE_ASYNC_LDS 0
#endif

// 16-byte global -> LDS copy: async DMA when available, else via VGPRs.
// Builtin signature (from hipcc diagnostic): (int4 AS1*, int4 AS3*, Ii, Ii).
__device__ __forceinline__ void copy16_to_lds(const float* g, float* l) {
#if HAVE_ASYNC_LDS
  __builtin_amdgcn_global_load_async_to_lds_b128(
      (__attribute__((address_space(1))) i4v*)(void*)g,
      (__attribute__((address_space(3))) i4v*)(void*)l, 0, 0);
#else
  *(f4v*)l = *(const f4v*)g;
#endif
}
__device__ __forceinline__ void async_wait_all() {
#if HAVE_ASYNC_LDS
#if __has_builtin(__builtin_amdgcn_s_wait_asynccnt)
  __builtin_amdgcn_s_wait_asynccnt(0);
#else
  asm volatile("s_wait_asynccnt 0" ::: "memory");
#endif
#endif
}

__device__ __forceinline__ v8f wmma4(v2f a, v2f b, v8f c) {
  // D = A(16x4,f32) x B(4x16,f32) + C(16x16,f32)
  return __builtin_amdgcn_wmma_f32_16x16x4_f32(false, a, false, b,
                                               (short)0, c, false, false);
}

// ---------------------------------------------------------------------------
// Kernel 1: complex LayerNorm over H=1024.  One block of 256 per (b,s) row.
// ---------------------------------------------------------------------------
__global__ __launch_bounds__(256)
void cln_kernel(const float* __restrict__ hidden,   // [2][M][H]
                const float* __restrict__ lng,      // [2][H]
                const float* __restrict__ lnb,      // [2][H]
                float* __restrict__ normed)         // [2][M][H]
{
  const int m    = blockIdx.x;
  const int tid  = threadIdx.x;
  const int lane = tid & 31, wv = tid >> 5;
  const float* xr = hidden + (size_t)m * Hn;
  const float* xi = hidden + (size_t)Mn * Hn + (size_t)m * Hn;

  float vr[4], vi[4];
  float sr = 0.f, si = 0.f;
#pragma unroll
  for (int j = 0; j < 4; ++j) {
    int h = tid + 256 * j;
    vr[j] = xr[h]; vi[j] = xi[h];
    sr += vr[j];   si += vi[j];
  }
#pragma unroll
  for (int o = 16; o > 0; o >>= 1) { sr += __shfl_xor(sr, o); si += __shfl_xor(si, o); }

  __shared__ float wr_[8], wi_[8];
  if (lane == 0) { wr_[wv] = sr; wi_[wv] = si; }
  __syncthreads();
  float tr = 0.f, ti = 0.f;
#pragma unroll
  for (int k = 0; k < 8; ++k) { tr += wr_[k]; ti += wi_[k]; }
  const float mur = tr * (1.0f / Hn), mui = ti * (1.0f / Hn);

  float sv = 0.f;
#pragma unroll
  for (int j = 0; j < 4; ++j) {
    float cr = vr[j] - mur, ci = vi[j] - mui;
    sv += cr * cr + ci * ci;
  }
#pragma unroll
  for (int o = 16; o > 0; o >>= 1) sv += __shfl_xor(sv, o);
  __syncthreads();
  if (lane == 0) wr_[wv] = sv;
  __syncthreads();
  float tv = 0.f;
#pragma unroll
  for (int k = 0; k < 8; ++k) tv += wr_[k];
  const float inv = rsqrtf(tv * (1.0f / Hn) + 1e-5f);

#pragma unroll
  for (int j = 0; j < 4; ++j) {
    int h = tid + 256 * j;
    float cr = (vr[j] - mur) * inv, ci = (vi[j] - mui) * inv;
    float gr = lng[h], gi = lng[Hn + h];
    normed[(size_t)m * Hn + h]                   = cr * gr - ci * gi + lnb[h];
    normed[(size_t)Mn * Hn + (size_t)m * Hn + h] = cr * gi + ci * gr + lnb[Hn + h];
  }
}

// ---------------------------------------------------------------------------
// Kernel 2: complex GEMM  out = A(z) @ W^T + bias,  M=4096, N=K=1024.
// Tile 128x64 per block (8 waves = 4x2, each wave 2x2 16x16 WMMA subtiles).
// A tiles staged by async global->LDS DMA (double-buffered); W tiles go
// through VGPRs into a pair-major LDS layout (every B fragment = b64 load).
// A row pitch 20: 16B-aligned DMA targets + conflict-free fragment reads.
// MODE 0: Q projection -> [2][M][H], scaled by 1/sqrt(HD)
// MODE 1: write into new_k/new_v at rows L..L+S of [2][B][KV][H]
// MODE 2: residual add (y = hidden + delta) into d_out [2][B][S][H]
// ---------------------------------------------------------------------------
template <int MODE>
__global__ __launch_bounds__(256)
void cgemm128x64(const float* __restrict__ Ar, const float* __restrict__ Ai,
                 const float* __restrict__ Wr, const float* __restrict__ Wi,
                 const float* __restrict__ biasR, const float* __restrict__ biasI,
                 float* __restrict__ outBase, const float* __restrict__ residBase)
{
  constexpr int K = Hn;
  const int tid  = threadIdx.x;
  const int lane = tid & 31, wave = tid >> 5;
  const int wm   = wave & 3, wn = wave >> 2;
  const int hi   = lane >> 4, ln16 = lane & 15;
  const int m0   = blockIdx.y * 128;
  const int n0   = blockIdx.x * 64;

  __shared__ __align__(16) float As[2][2][128][20];    // [buf][re/im][m][k]
  __shared__ __align__(16) float BsP[2][2][8][64][2];  // [buf][re/im][k/2][n][pair]

  v8f accR[2][2], accI[2][2];
#pragma unroll
  for (int a = 0; a < 2; ++a)
#pragma unroll
    for (int b = 0; b < 2; ++b) { accR[a][b] = {}; accI[a][b] = {}; }

  const int lr = tid >> 2;          // 0..63 : A row (and +64), W row n
  const int lc = (tid & 3) * 4;     // 0,4,8,12 : k column group

  auto asyncA = [&](int buf, int k0) {
    const float* pr = Ar + (size_t)(m0 + lr) * K + k0 + lc;
    const float* pi = Ai + (size_t)(m0 + lr) * K + k0 + lc;
    copy16_to_lds(pr,                  &As[buf][0][lr][lc]);
    copy16_to_lds(pr + (size_t)64 * K, &As[buf][0][lr + 64][lc]);
    copy16_to_lds(pi,                  &As[buf][1][lr][lc]);
    copy16_to_lds(pi + (size_t)64 * K, &As[buf][1][lr + 64][lc]);
  };

  f4v ldbR, ldbI;
  auto fetchB = [&](int k0) {
    ldbR = *(const f4v*)(Wr + (size_t)(n0 + lr) * K + k0 + lc);
    ldbI = *(const f4v*)(Wi + (size_t)(n0 + lr) * K + k0 + lc);
  };
  auto storeB = [&](int buf) {
    const int kp = lc >> 1;                       // 0,2,4,6
    *(v2f*)&BsP[buf][0][kp][lr][0]     = (v2f){ldbR[0], ldbR[1]};
    *(v2f*)&BsP[buf][0][kp + 1][lr][0] = (v2f){ldbR[2], ldbR[3]};
    *(v2f*)&BsP[buf][1][kp][lr][0]     = (v2f){ldbI[0], ldbI[1]};
    *(v2f*)&BsP[buf][1][kp + 1][lr][0] = (v2f){ldbI[2], ldbI[3]};
  };

  asyncA(0, 0);
  fetchB(0);
  storeB(0);
  async_wait_all();
  __syncthreads();

  constexpr int NSTAGE = K / 16;   // 64
  for (int s = 0; s < NSTAGE; ++s) {
    const int  buf  = s & 1;
    const bool more = (s + 1 < NSTAGE);
    if (more) { asyncA(buf ^ 1, (s + 1) * 16); fetchB((s + 1) * 16); }
    if (s + 2 < NSTAGE) {   // L2 prefetch hint for the tile after next
      __builtin_prefetch(Ar + (size_t)(m0 + lr) * K + (s + 2) * 16, 0, 0);
      __builtin_prefetch(Wr + (size_t)(n0 + lr) * K + (s + 2) * 16, 0, 0);
    }
#pragma unroll
    for (int kk = 0; kk < 16; kk += 4) {
      const int kp = (kk >> 1) + hi;             // pair row for this half-wave
      v2f aR[2], aI[2], naI[2], bR[2], bI[2];
#pragma unroll
      for (int mi = 0; mi < 2; ++mi) {
        const int mr = wm * 32 + mi * 16 + ln16;
        aR[mi]  = *(const v2f*)&As[buf][0][mr][kk + 2 * hi];  // aligned b64
        aI[mi]  = *(const v2f*)&As[buf][1][mr][kk + 2 * hi];
        naI[mi] = -aI[mi];
      }
#pragma unroll
      for (int ni = 0; ni < 2; ++ni) {
        const int nc = wn * 32 + ni * 16 + ln16;
        bR[ni] = *(const v2f*)&BsP[buf][0][kp][nc][0];        // aligned b64
        bI[ni] = *(const v2f*)&BsP[buf][1][kp][nc][0];
      }
#pragma unroll
      for (int mi = 0; mi < 2; ++mi)
#pragma unroll
        for (int ni = 0; ni < 2; ++ni) {
          accR[mi][ni] = wmma4(aR[mi],  bR[ni], accR[mi][ni]);  // zr*wr
          accR[mi][ni] = wmma4(naI[mi], bI[ni], accR[mi][ni]);  // -zi*wi
          accI[mi][ni] = wmma4(aR[mi],  bI[ni], accI[mi][ni]);  // zr*wi
          accI[mi][ni] = wmma4(aI[mi],  bR[ni], accI[mi][ni]);  // zi*wr
        }
    }
    if (more) storeB(buf ^ 1);
    async_wait_all();     // this iteration's DMA (if any) has landed
    __syncthreads();
  }

  // Epilogue: bias, routing, optional residual / score pre-scale.
#pragma unroll
  for (int mi = 0; mi < 2; ++mi)
#pragma unroll
    for (int ni = 0; ni < 2; ++ni) {
      const int   gcol = n0 + wn * 32 + ni * 16 + ln16;
      const float br   = biasR[gcol], bi = biasI[gcol];
#pragma unroll
      for (int v = 0; v < 8; ++v) {
        const int gm = m0 + wm * 32 + mi * 16 + v + 8 * hi;   // C layout row
        float vr = accR[mi][ni][v] + br;
        float vi = accI[mi][ni][v] + bi;
        if constexpr (MODE == 0) {       // fold 1/sqrt(HD) into Q itself
          vr *= 0.125f; vi *= 0.125f;
        }
        size_t offR, offI;
        if constexpr (MODE == 1) {
          const int bb = gm >> 11;            // / Sn
          const int ss = gm & (Sn - 1);
          offR = ((size_t)bb * KVn + (size_t)(Ln + ss)) * Hn + gcol;
          offI = offR + (size_t)Bn * KVn * Hn;
        } else {
          offR = (size_t)gm * Hn + gcol;
          offI = offR + (size_t)Mn * Hn;
        }
        if constexpr (MODE == 2) {
          vr += residBase[offR];
          vi += residBase[offI];
        }
        outBase[offR] = vr;
        outBase[offI] = vi;
      }
    }
}

// ---------------------------------------------------------------------------
// Kernel 3: copy k_cache/v_cache [2][B][L][H] into rows 0..L of [2][B][KV][H].
// ---------------------------------------------------------------------------
__global__ __launch_bounds__(256)
void cache_copy(const float* __restrict__ src, float* __restrict__ dst)
{
  const size_t total4 = (size_t)2 * Bn * Ln * Hn / 4;
  size_t idx = (size_t)blockIdx.x * blockDim.x + threadIdx.x;
  if (idx >= total4) return;
  const size_t lin   = idx * 4;
  const size_t chunk = (size_t)Ln * Hn;       // contiguous per (plane,b)
  const size_t cb    = lin / chunk;
  const size_t rem   = lin - cb * chunk;
  f4v v = *(const f4v*)(src + lin);
  *(f4v*)(dst + cb * (size_t)KVn * Hn + rem) = v;
}

// ---------------------------------------------------------------------------
// Kernel 4: flash attention.  Block = 64 queries x (head, batch), 4 waves;
// wave w owns queries [w*16, w*16+16).  kv in 32-key blocks; Q LDS-resident
// (staged once via async DMA; pre-scaled in the Q projection).
// kv loop split: phase A (cache prefix, no mask code) and phase B (causal
// region, per-element mask) stopping at the last partially-visible block --
// fully-masked blocks have exactly-zero softmax weight and are skipped.
// K^T and V time-share one pair-major LDS buffer (every B fragment = b64).
// ---------------------------------------------------------------------------
__global__ __launch_bounds__(128)
void flash_attn(const float* __restrict__ Q,    // [2][M][H] scratch (pre-scaled)
                const float* __restrict__ Kc,   // [2][B][KV][H] (= new_k)
                const float* __restrict__ Vc,   // [2][B][KV][H] (= new_v)
                float* __restrict__ O)          // [2][M][H] scratch
{
  const int qt   = blockIdx.x;          // query tile (64)
  const int head = blockIdx.y;
  const int b    = blockIdx.z;
  const int tid  = threadIdx.x, lane = tid & 31, w = tid >> 5;
  const int hi   = lane >> 4, ln16 = lane & 15;
  const int q0   = qt * 64;

  // LDS (CU-mode safe, < 64KB):
  //   Qs    2*64*68*4   = 34.8 KB (persistent; pitch 68 -> 16B-aligned DMA)
  //   KVraw 2*32*32*2*4 = 16.0 KB (K^T phase [p][d/2][key][2];
  //                                V  phase [p][key/2][d][2], aliased)
  //   Ps    4*16*34*4   =  8.7 KB (per-wave P' C-layout -> A-layout bridge)
  __shared__ __align__(16) float Qs[2][64][68];
  __shared__ __align__(16) float KVraw[2][32][32][2];
  __shared__ __align__(16) float Ps[4][16][34];

  const size_t QPS   = (size_t)Mn * Hn;
  const size_t KPS   = (size_t)Bn * KVn * Hn;
  const size_t kbase = (size_t)b * KVn * Hn + (size_t)head * HD;

  // ---- stage Q via async DMA (pure copy; scale already applied)
  for (int i = tid; i < 2048; i += 128) {     // 2 planes * 64 rows * 16 f4
    const int p = i >> 10, r = (i >> 4) & 63, c4 = (i & 15) << 2;
    copy16_to_lds(Q + (size_t)p * QPS +
                  (size_t)(b * Sn + q0 + r) * Hn + head * HD + c4,
                  &Qs[p][r][c4]);
  }
  async_wait_all();
  __syncthreads();

  v8f oR[4], oI[4];
#pragma unroll
  for (int t = 0; t < 4; ++t) { oR[t] = {}; oI[t] = {}; }
  float mst[8], lst[8];
#pragma unroll
  for (int v = 0; v < 8; ++v) { mst[v] = -3.0e38f; lst[v] = 0.f; }

  const int qrow = w * 16 + ln16;
  float* const Vbuf = (float*)KVraw;   // V view: [p][key/2][d][2]

  auto kv_block = [&](int kb, bool domask) {
    // ---- stage K^T pairs : KVraw[p][d/2][key][2]
    for (int i = tid; i < 1024; i += 128) {   // 2 planes * 32 keys * 16 f4
      const int p = i >> 9, key = (i >> 4) & 31, c4 = (i & 15) << 2;
      f4v v = *(const f4v*)(Kc + (size_t)p * KPS + kbase +
                            (size_t)(kb * 32 + key) * Hn + c4);
      *(v2f*)&KVraw[p][(c4 >> 1)][key][0]     = (v2f){v[0], v[1]};
      *(v2f*)&KVraw[p][(c4 >> 1) + 1][key][0] = (v2f){v[2], v[3]};
    }
    if (kb + 1 < KVn / 32)   // L2 prefetch hint for next key block
      __builtin_prefetch(Kc + kbase + (size_t)((kb + 1) * 32 + (tid >> 2)) * Hn, 0, 0);
    __syncthreads();

    // ---- GEMM1: P = Qr*Kr^T + Qi*Ki^T  (Hermitian real part, pre-scaled)
    v8f P[2];
    P[0] = {}; P[1] = {};
#pragma unroll
    for (int st = 0; st < 16; ++st) {
      const int k  = st * 4 + 2 * hi;
      const int dp = st * 2 + hi;
      v2f ar = *(const v2f*)&Qs[0][qrow][k];   // aligned b64
      v2f ai = *(const v2f*)&Qs[1][qrow][k];
#pragma unroll
      for (int nt = 0; nt < 2; ++nt) {
        const int nc = nt * 16 + ln16;
        v2f br = *(const v2f*)&KVraw[0][dp][nc][0];   // aligned b64
        v2f bi = *(const v2f*)&KVraw[1][dp][nc][0];
        P[nt] = wmma4(ar, br, P[nt]);
        P[nt] = wmma4(ai, bi, P[nt]);
      }
    }

    // ---- (optional) causal mask + online softmax
    float rmax[8];
#pragma unroll
    for (int v = 0; v < 8; ++v) rmax[v] = -3.0e38f;
#pragma unroll
    for (int nt = 0; nt < 2; ++nt) {
      const int key_g = kb * 32 + nt * 16 + ln16;
#pragma unroll
      for (int v = 0; v < 8; ++v) {
        float x = P[nt][v];
        if (domask) {
          const int q_g = q0 + w * 16 + v + 8 * hi;
          if (key_g - Ln > q_g) x = -3.0e38f;
          P[nt][v] = x;
        }
        rmax[v] = fmaxf(rmax[v], x);
      }
    }
#pragma unroll
    for (int v = 0; v < 8; ++v) {
#pragma unroll
      for (int o = 1; o < 16; o <<= 1) rmax[v] = fmaxf(rmax[v], __shfl_xor(rmax[v], o));
      const float mn = fmaxf(mst[v], rmax[v]);
      const float sc = __expf(mst[v] - mn);
      float s = 0.f;
#pragma unroll
      for (int nt = 0; nt < 2; ++nt) {
        const float e = __expf(P[nt][v] - mn);
        s += e;
        Ps[w][v + 8 * hi][nt * 16 + ln16] = e;    // P' in [m][k] for GEMM2 A
      }
#pragma unroll
      for (int o = 1; o < 16; o <<= 1) s += __shfl_xor(s, o);
      lst[v] = lst[v] * sc + s;
      mst[v] = mn;
#pragma unroll
      for (int nt = 0; nt < 4; ++nt) { oR[nt][v] *= sc; oI[nt][v] *= sc; }
    }
    __syncthreads();   // all waves done with KVraw(K^T)

    // ---- stage V pairs : Vbuf[p][key/2][d][2]
    for (int i = tid; i < 512; i += 128) {    // 2 planes * 16 keypairs * 16 f4
      const int p = i >> 8, kp2 = (i >> 4) & 15, c4 = (i & 15) << 2;
      const float* vp = Vc + (size_t)p * KPS + kbase +
                        (size_t)(kb * 32 + kp2 * 2) * Hn + c4;
      f4v v0 = *(const f4v*)vp;
      f4v v1 = *(const f4v*)(vp + Hn);
#pragma unroll
      for (int e = 0; e < 4; ++e)
        *(v2f*)&Vbuf[(size_t)p * 2048 + kp2 * 128 + (c4 + e) * 2] = (v2f){v0[e], v1[e]};
    }
    __syncthreads();

    // ---- GEMM2: out += P' @ Vr , P' @ Vi   (k = 32 keys -> 8 steps)
#pragma unroll
    for (int st = 0; st < 8; ++st) {
      const int k  = st * 4 + 2 * hi;
      const int kp = st * 2 + hi;
      v2f a = *(const v2f*)&Ps[w][ln16][k];    // aligned b64
#pragma unroll
      for (int nt = 0; nt < 4; ++nt) {
        const int nc = nt * 16 + ln16;
        v2f bvr = *(const v2f*)&Vbuf[0 * 2048 + kp * 128 + nc * 2];   // aligned b64
        v2f bvi = *(const v2f*)&Vbuf[1 * 2048 + kp * 128 + nc * 2];
        oR[nt] = wmma4(a, bvr, oR[nt]);
        oI[nt] = wmma4(a, bvi, oI[nt]);
      }
    }
    __syncthreads();   // before next K^T staging overwrites KVraw
  };

  // Phase A: cache prefix, never masked.
  for (int kb = 0; kb < Ln / 32; ++kb) kv_block(kb, false);
  // Phase B: causal region; stop after the last partially-visible block.
  const int kbEnd = (Ln + q0 + 64 + 31) / 32;   // fully-masked blocks skipped
  for (int kb = Ln / 32; kb < kbEnd; ++kb) kv_block(kb, true);

  // ---- normalize and write [2][M][H]
#pragma unroll
  for (int nt = 0; nt < 4; ++nt)
#pragma unroll
    for (int v = 0; v < 8; ++v) {
      const int q_g = q0 + w * 16 + v + 8 * hi;
      const float inv = 1.0f / lst[v];
      const size_t off = (size_t)(b * Sn + q_g) * Hn + head * HD + nt * 16 + ln16;
      O[off]       = oR[nt][v] * inv;
      O[QPS + off] = oI[nt][v] * inv;
    }
}

// ---------------------------------------------------------------------------
extern "C" void kernel_launch(void* const* d_in, const int* in_sizes, int n_in,
                              void* d_out, int out_size, void* d_ws, size_t ws_size,
                              hipStream_t stream) {
  (void)in_sizes; (void)n_in; (void)out_size; (void)ws_size;
  const float* hidden  = (const float*)d_in[0];
  const float* k_cache = (const float*)d_in[1];
  const float* v_cache = (const float*)d_in[2];
  const float* qw = (const float*)d_in[3];
  const float* qb = (const float*)d_in[4];
  const float* kw = (const float*)d_in[5];
  const float* kb = (const float*)d_in[6];
  const float* vw = (const float*)d_in[7];
  const float* vb = (const float*)d_in[8];
  const float* ow = (const float*)d_in[9];
  const float* ob = (const float*)d_in[10];
  const float* lng = (const float*)d_in[11];
  const float* lnb = (const float*)d_in[12];

  float* out   = (float*)d_out;                            // [2][B][S][H]
  float* new_k = out + (size_t)2 * Bn * Sn * Hn;           // [2][B][KV][H]
  float* new_v = new_k + (size_t)2 * Bn * KVn * Hn;        // [2][B][KV][H]

  float* ws     = (float*)d_ws;
  float* normed = ws;                                      // [2][M][H]
  float* Qbuf   = ws + (size_t)2 * Mn * Hn;                // [2][M][H]
  float* attn   = Qbuf + (size_t)2 * Mn * Hn;              // [2][M][H]

  const size_t WW = (size_t)Hn * Hn;   // weight plane stride

  // 1) complex layernorm
  cln_kernel<<<Mn, 256, 0, stream>>>(hidden, lng, lnb, normed);

  // 2) Q/K/V complex projections (K/V land directly in the concatenated cache)
  dim3 gg(Hn / 64, Mn / 128);
  cgemm128x64<0><<<gg, 256, 0, stream>>>(normed, normed + (size_t)Mn * Hn,
                                         qw, qw + WW, qb, qb + Hn, Qbuf, nullptr);
  cgemm128x64<1><<<gg, 256, 0, stream>>>(normed, normed + (size_t)Mn * Hn,
                                         kw, kw + WW, kb, kb + Hn, new_k, nullptr);
  cgemm128x64<1><<<gg, 256, 0, stream>>>(normed, normed + (size_t)Mn * Hn,
                                         vw, vw + WW, vb, vb + Hn, new_v, nullptr);

  // 3) prepend caches (rows 0..L of new_k/new_v)
  const size_t tot4 = (size_t)2 * Bn * Ln * Hn / 4;
  const int cblocks = (int)((tot4 + 255) / 256);
  cache_copy<<<cblocks, 256, 0, stream>>>(k_cache, new_k);
  cache_copy<<<cblocks, 256, 0, stream>>>(v_cache, new_v);

  // 4) flash attention over the full kv window
  flash_attn<<<dim3(Sn / 64, NH, Bn), 128, 0, stream>>>(Qbuf, new_k, new_v, attn);

  // 5) output projection + residual
  cgemm128x64<2><<<gg, 256, 0, stream>>>(attn, attn + (size_t)Mn * Hn,
                                         ow, ow + WW, ob, ob + Hn, out, hidden);
}